// LJC_17454747090959
// MI455X (gfx1250) — compile-verified
//
#include <hip/hip_runtime.h>
#include <math.h>

// ---------------- fused MGCN block for MI455X (gfx1250, wave32) --------------
// One kernel, all intermediates in LDS. GEMMs use V_WMMA_F32_16X16X4_F32.

typedef float v8f __attribute__((ext_vector_type(8)));
typedef float v2f __attribute__((ext_vector_type(2)));

#define NJ       17
#define TB       4
#define MROWS    (TB * NJ)        // 68 real rows per workgroup
#define MTILES   5
#define MPAD     (MTILES * 16)    // 80 padded rows
#define DIMC     512
#define HIDC     384
#define ASTR     516              // x / h2 LDS stride (4*row % 64 bank spread)
#define GSTR     388              // g LDS stride
#define HBS      34               // h0/h1 partial buffer stride
#define NCH      32               // N-chunk width per mix pass
#define NTHR     256              // 8 waves

__device__ __forceinline__ float wsum(float v) {
#pragma unroll
    for (int o = 16; o > 0; o >>= 1) v += __shfl_xor(v, o, 32);
    return v;
}

// One GEMM (+K-split combine) + graph-mix pass.
// Wg: [2][KD][ND] row major, Mg: [17][ND], biasg: [ND].
// sAin (stride AS) -> sOut (stride OS). h0 = j0 GEMM, h1 = j1 GEMM.
template <int KD, int ND, int AS, int OS>
__device__ void phase(const float* __restrict__ Wg, const float* __restrict__ Mg,
                      const float* __restrict__ biasg, const float* sAdj,
                      const float* sAin, float* sOut,
                      float* sH, float* sMc, float* sBias, bool doGelu) {
    const int t    = threadIdx.x;
    const int wave = t >> 5, lane = t & 31;
    const int j    = wave & 1;            // which weight matrix
    const int ct   = (wave >> 1) & 1;     // 16-col tile within 32-col chunk
    const int rg   = wave >> 2;           // K-half
    const int laneN = lane & 15;
    const int kOff  = (lane >> 4) << 1;   // 0 or 2 (A/B half-lane K offset)
    const int k0 = rg * (KD / 2), k1 = k0 + KD / 2;
    const float* W = Wg + j * KD * ND;

    for (int nc = 0; nc < ND / NCH; ++nc) {
        const int nb = nc * NCH + ct * 16;
        v8f acc[MTILES];
#pragma unroll
        for (int rt = 0; rt < MTILES; ++rt)
#pragma unroll
            for (int g = 0; g < 8; ++g) acc[rt][g] = 0.0f;

#pragma unroll 4
        for (int kb = k0; kb < k1; kb += 4) {
            const int kr = kb + kOff;
            // B frag (4x16): v0 = row kr, v1 = row kr+1, col = nb+laneN
            v2f b;
            b.x = W[kr * ND + nb + laneN];
            b.y = W[(kr + 1) * ND + nb + laneN];
#pragma unroll
            for (int rt = 0; rt < MTILES; ++rt) {
                // A frag (16x4): lane M = laneN, K = {kr, kr+1}
                v2f a = *(const v2f*)(sAin + (rt * 16 + laneN) * AS + kr);
                acc[rt] = __builtin_amdgcn_wmma_f32_16x16x4_f32(
                    false, a, false, b, (short)0, acc[rt], false, false);
            }
        }
        // combine K-halves through LDS: rg1 stores, rg0 adds
        if (rg == 1) {
#pragma unroll
            for (int rt = 0; rt < MTILES; ++rt)
#pragma unroll
                for (int g = 0; g < 8; ++g)
                    sH[(j * MPAD + rt * 16 + ((lane >> 4) << 3) + g) * HBS +
                       ct * 16 + laneN] = acc[rt][g];
        }
        __syncthreads();
        if (rg == 0) {
#pragma unroll
            for (int rt = 0; rt < MTILES; ++rt)
#pragma unroll
                for (int g = 0; g < 8; ++g) {
                    int o = (j * MPAD + rt * 16 + ((lane >> 4) << 3) + g) * HBS +
                            ct * 16 + laneN;
                    sH[o] += acc[rt][g];
                }
        } else {  // rg1 waves stage M/bias chunk meanwhile
            int tt = (wave - 4) * 32 + lane;  // 0..127
            for (int idx = tt; idx < NJ * NCH; idx += 128) {
                int w2 = idx / NCH, d = idx % NCH;
                sMc[idx] = Mg[w2 * ND + nc * NCH + d];
            }
            if (tt < NCH) sBias[tt] = biasg[nc * NCH + tt];
        }
        __syncthreads();
        // graph mix: out[v,d] = A[v,v]*M[v,d]*h0[v,d] + sum_{w!=v} A[v,w]*M[w,d]*h1[w,d]
        for (int idx = t; idx < MROWS * NCH; idx += NTHR) {
            int r = idx >> 5, d = idx & 31;   // wave-uniform r, d = lane
            int v = r % NJ, rb = r - v;
            float a0 = sAdj[v * NJ + v] * sMc[v * NCH + d] * sH[r * HBS + d];
#pragma unroll
            for (int w2 = 0; w2 < NJ; ++w2) {
                if (w2 == v) continue;
                a0 += sAdj[v * NJ + w2] * sMc[w2 * NCH + d] *
                      sH[(MPAD + rb + w2) * HBS + d];
            }
            a0 += sBias[d];
            if (doGelu) a0 = 0.5f * a0 * (1.0f + erff(a0 * 0.70710678118654752f));
            sOut[r * OS + nc * NCH + d] = a0;
        }
        __syncthreads();
    }
}

__global__ void __launch_bounds__(NTHR, 1)
fused_mgcn(const float* __restrict__ x, const float* __restrict__ bone,
           const float* __restrict__ adjb, const float* __restrict__ g1,
           const float* __restrict__ b1, const float* __restrict__ W1,
           const float* __restrict__ M1, const float* __restrict__ A21,
           const float* __restrict__ bias1, const float* __restrict__ W2,
           const float* __restrict__ M2, const float* __restrict__ A22,
           const float* __restrict__ bias2, const float* __restrict__ g2,
           const float* __restrict__ b2, float* __restrict__ out) {
    __shared__ float sA[MPAD * ASTR];        // x(normalized) then h2
    __shared__ float sG[MPAD * GSTR];        // gelu(mix1) = A of GEMM2
    __shared__ float sH[2 * MPAD * HBS];     // h0/h1 chunk
    __shared__ float sMc[NJ * NCH];
    __shared__ float sAdj1[NJ * NJ], sAdj2[NJ * NJ];
    __shared__ float sBias[NCH];

    const int t = threadIdx.x;
    const int wave = t >> 5, lane = t & 31;
    const long long base = (long long)blockIdx.x * (TB * NJ) * DIMC;
    const float* xg = x + base;

    // stage x tile; zero pad rows of A and G
    for (int idx = t; idx < MROWS * DIMC; idx += NTHR) {
        int r = idx >> 9, c = idx & (DIMC - 1);
        sA[r * ASTR + c] = xg[idx];
    }
    for (int idx = t; idx < (MPAD - MROWS) * DIMC; idx += NTHR) {
        int r = MROWS + (idx >> 9), c = idx & (DIMC - 1);
        sA[r * ASTR + c] = 0.0f;
    }
    for (int idx = t; idx < (MPAD - MROWS) * HIDC; idx += NTHR) {
        int r = MROWS + idx / HIDC, c = idx % HIDC;
        sG[r * GSTR + c] = 0.0f;
    }
    // symmetrized adjacencies: 0.5*((adj+A2) + (adj+A2)^T)
    for (int idx = t; idx < NJ * NJ; idx += NTHR) {
        int v = idx / NJ, w = idx % NJ;
        float f = adjb[v * NJ + w] + adjb[w * NJ + v];
        sAdj1[idx] = 0.5f * (f + A21[v * NJ + w] + A21[w * NJ + v]);
        sAdj2[idx] = 0.5f * (f + A22[v * NJ + w] + A22[w * NJ + v]);
    }
    __syncthreads();

    // LayerNorm 1 in place (one wave per row)
    for (int r = wave; r < MROWS; r += 8) {
        float s = 0.0f, q = 0.0f;
#pragma unroll
        for (int i = 0; i < 16; ++i) {
            float v = sA[r * ASTR + lane + (i << 5)];
            s += v; q += v * v;
        }
        s = wsum(s); q = wsum(q);
        float m  = s * (1.0f / DIMC);
        float rs = rsqrtf(q * (1.0f / DIMC) - m * m + 1e-5f);
#pragma unroll
        for (int i = 0; i < 16; ++i) {
            int c = lane + (i << 5);
            sA[r * ASTR + c] = (sA[r * ASTR + c] - m) * rs * g1[c] + b1[c];
        }
    }
    __syncthreads();

    phase<DIMC, HIDC, ASTR, GSTR>(W1, M1, bias1, sAdj1, sA, sG, sH, sMc, sBias, true);
    phase<HIDC, DIMC, GSTR, ASTR>(W2, M2, bias2, sAdj2, sG, sA, sH, sMc, sBias, false);

    // LN2 + residual + bone gate
    for (int r = wave; r < MROWS; r += 8) {
        float s = 0.0f, q = 0.0f;
#pragma unroll
        for (int i = 0; i < 16; ++i) {
            float v = sA[r * ASTR + lane + (i << 5)];
            s += v; q += v * v;
        }
        int v = r % NJ, bl = r / NJ;
        long long bidx = ((long long)(blockIdx.x * TB + bl) * NJ + v);
        float bsum = bone[bidx * 192 + lane] + bone[bidx * 192 + 32 + lane];
        s = wsum(s); q = wsum(q); bsum = wsum(bsum);
        float m  = s * (1.0f / DIMC);
        float rs = rsqrtf(q * (1.0f / DIMC) - m * m + 1e-5f);
        float gate  = 1.0f / (1.0f + expf(-0.01f * (bsum * (1.0f / 64.0f))));
        float scale = 1.0f +
            ((v == 3 || v == 6 || v == 10 || v == 13 || v == 16) ? 0.03f * gate : 0.0f);
#pragma unroll
        for (int i = 0; i < 16; ++i) {
            int c = lane + (i << 5);
            float h = sA[r * ASTR + c];
            out[base + (long long)r * DIMC + c] =
                (xg[(long long)r * DIMC + c] + ((h - m) * rs * g2[c] + b2[c])) * scale;
        }
    }
}

extern "C" void kernel_launch(void* const* d_in, const int* in_sizes, int n_in,
                              void* d_out, int out_size, void* d_ws, size_t ws_size,
                              hipStream_t stream) {
    const float* x     = (const float*)d_in[0];
    const float* bone  = (const float*)d_in[1];
    const float* adjb  = (const float*)d_in[2];
    const float* g1    = (const float*)d_in[3];
    const float* b1    = (const float*)d_in[4];
    const float* W1    = (const float*)d_in[5];
    const float* M1    = (const float*)d_in[6];
    const float* A21   = (const float*)d_in[7];
    const float* bias1 = (const float*)d_in[8];
    const float* W2    = (const float*)d_in[9];
    const float* M2    = (const float*)d_in[10];
    const float* A22   = (const float*)d_in[11];
    const float* bias2 = (const float*)d_in[12];
    const float* g2    = (const float*)d_in[13];
    const float* b2    = (const float*)d_in[14];

    int B = in_sizes[0] / (NJ * DIMC);      // 8192
    int grid = B / TB;                       // 2048 (B divisible by TB)
    fused_mgcn<<<grid, NTHR, 0, stream>>>(x, bone, adjb, g1, b1, W1, M1, A21,
                                          bias1, W2, M2, A22, bias2, g2, b2,
                                          (float*)d_out);
}